// LinkPredictionModel_69234872812250
// MI455X (gfx1250) — compile-verified
//
#include <hip/hip_runtime.h>

typedef __attribute__((ext_vector_type(2))) float v2f;
typedef __attribute__((ext_vector_type(8))) float v8f;

// ---------------------------------------------------------------- zero fill
__global__ void zero_f32(float* __restrict__ p, long n) {
  long i = (long)blockIdx.x * blockDim.x + threadIdx.x;
  long stride = (long)gridDim.x * blockDim.x;
  for (; i < n; i += stride) p[i] = 0.0f;
}

// ---------------------------------------------------------------- degree
__global__ void deg_kernel(const int* __restrict__ dst, float* __restrict__ deg, int E) {
  int e = blockIdx.x * blockDim.x + threadIdx.x;
  if (e < E) atomicAdd(&deg[dst[e]], 1.0f);
}

// ---------------------------------------------------------------- scatter add: agg[dst] += feat[src]
__global__ void scatter_kernel(const float* __restrict__ feat,
                               const int* __restrict__ src,
                               const int* __restrict__ dst,
                               float* __restrict__ agg, int E, int D) {
  long i = (long)blockIdx.x * blockDim.x + threadIdx.x;
  long total = (long)E * D;
  if (i >= total) return;
  int e = (int)(i / D);
  int f = (int)(i - (long)e * D);
  atomicAdd(&agg[(long)dst[e] * D + f], feat[(long)src[e] * D + f]);
}

// ---------------------------------------------------------------- SAGE layer GEMM via fp32 WMMA
// out[N,C] = act( (agg/deg) @ Wl + xin @ Wr + bias )
// One wave computes one 16x16 output tile using V_WMMA_F32_16X16X4_F32.
__global__ __launch_bounds__(256)
void sage_gemm_wmma(const float* __restrict__ agg,   // (N,K) neighbor sums
                    const float* __restrict__ deg,   // (N)
                    const float* __restrict__ xin,   // (N,K) self features
                    const float* __restrict__ Wl,    // (K,C)
                    const float* __restrict__ Wr,    // (K,C)
                    const float* __restrict__ bias,  // (C)
                    float* __restrict__ out,         // (N,C)
                    int N, int K, int C, int relu)
{
  const int lane = threadIdx.x & 31;
  const int wave = threadIdx.x >> 5;
  const int ct   = blockIdx.y * 8 + wave;      // column tile (wave-uniform)
  if (ct * 16 >= C) return;                    // uniform exit keeps EXEC all-1s for WMMA
  const int c0 = ct * 16;
  const int r0 = blockIdx.x * 16;

  const int m  = lane & 15;                    // row within A tile / col within C tile
  const int kg = lane >> 4;                    // K-group: lanes 0-15 -> K{0,1}, 16-31 -> K{2,3}
  int r = r0 + m; if (r > N - 1) r = N - 1;    // N % 16 == 0 in practice; clamp stays convergent

  const float d   = deg[r];
  const float inv = 1.0f / fmaxf(d, 1.0f);     // mean = agg / clip(deg, 1)

  v8f acc = {};
  for (int k = 0; k < K; k += 4) {
    // A tiles: 16x4 f32 layout -> per lane 2 consecutive K elements of row m
    const long arow = (long)r * K + k + 2 * kg;
    v2f am, ax;
    am.x = agg[arow] * inv;  am.y = agg[arow + 1] * inv;
    ax.x = xin[arow];        ax.y = xin[arow + 1];
    // B tiles: 4x16 f32; VGPR0 carries rows k+{0,1}, VGPR1 rows k+{2,3}, cols striped over lanes
    const long brow0 = (long)(k + kg)     * C + c0 + m;
    const long brow1 = (long)(k + 2 + kg) * C + c0 + m;
    v2f bl, br;
    bl.x = Wl[brow0]; bl.y = Wl[brow1];
    br.x = Wr[brow0]; br.y = Wr[brow1];
    // D = A*B + C, exact fp32 (matches reference numerics)
    acc = __builtin_amdgcn_wmma_f32_16x16x4_f32(false, am, false, bl, (short)0, acc, false, false);
    acc = __builtin_amdgcn_wmma_f32_16x16x4_f32(false, ax, false, br, (short)0, acc, false, false);
  }

  const float bv = bias[c0 + m];
#pragma unroll
  for (int i = 0; i < 8; ++i) {
    float v = acc[i] + bv;
    if (relu) v = fmaxf(v, 0.0f);
    int row = r0 + i + 8 * kg;                 // C/D layout: VGPR i, lane half -> row i / i+8
    if (row < N) out[(long)row * C + c0 + m] = v;
  }
}

// ---------------------------------------------------------------- edge dot decoder
__global__ void decode_kernel(const float* __restrict__ z,
                              const int* __restrict__ ea,
                              const int* __restrict__ eb,
                              float* __restrict__ out, int EP, int C) {
  int e = blockIdx.x * blockDim.x + threadIdx.x;
  if (e >= EP) return;
  const float4* za = (const float4*)(z + (long)ea[e] * C);
  const float4* zb = (const float4*)(z + (long)eb[e] * C);
  float s = 0.0f;
  for (int i = 0; i < C / 4; ++i) {
    float4 a = za[i], b = zb[i];
    s += a.x * b.x + a.y * b.y + a.z * b.z + a.w * b.w;
  }
  out[e] = s;
}

// ---------------------------------------------------------------- launcher
extern "C" void kernel_launch(void* const* d_in, const int* in_sizes, int n_in,
                              void* d_out, int out_size, void* d_ws, size_t ws_size,
                              hipStream_t stream) {
  const float* x   = (const float*)d_in[0];
  const int*   ei  = (const int*)d_in[1];
  const int*   pei = (const int*)d_in[2];
  const int*   nei = (const int*)d_in[3];
  const float* Wl1 = (const float*)d_in[4];
  const float* Wr1 = (const float*)d_in[5];
  const float* b1  = (const float*)d_in[6];
  const float* Wl2 = (const float*)d_in[7];
  const float* Wr2 = (const float*)d_in[8];
  const float* b2  = (const float*)d_in[9];
  float* out = (float*)d_out;

  const int IN_C = 128, HID = 128, OUTC = 64;
  const int N  = in_sizes[0] / IN_C;
  const int E  = in_sizes[1] / 2;
  const int EP = in_sizes[2] / 2;

  float* deg = (float*)d_ws;
  float* agg = deg + N;
  float* h   = agg + (size_t)N * HID;
  float* z   = h   + (size_t)N * HID;

  const int* src = ei;
  const int* dst = ei + E;

  const int T = 256;
  const long nz0 = (long)N * (HID + 1);           // deg + agg contiguous
  zero_f32<<<4096, T, 0, stream>>>(deg, nz0);
  deg_kernel<<<(E + T - 1) / T, T, 0, stream>>>(dst, deg, E);

  // ---- layer 1
  long sc1 = (long)E * IN_C;
  scatter_kernel<<<(int)((sc1 + T - 1) / T), T, 0, stream>>>(x, src, dst, agg, E, IN_C);
  sage_gemm_wmma<<<dim3((N + 15) / 16, 1), T, 0, stream>>>(agg, deg, x, Wl1, Wr1, b1, h,
                                                           N, IN_C, HID, 1);
  // ---- layer 2
  zero_f32<<<4096, T, 0, stream>>>(agg, (long)N * HID);
  long sc2 = (long)E * HID;
  scatter_kernel<<<(int)((sc2 + T - 1) / T), T, 0, stream>>>(h, src, dst, agg, E, HID);
  sage_gemm_wmma<<<dim3((N + 15) / 16, 1), T, 0, stream>>>(agg, deg, h, Wl2, Wr2, b2, z,
                                                           N, HID, OUTC, 0);
  // ---- decode: pos then neg, concatenated
  decode_kernel<<<(EP + T - 1) / T, T, 0, stream>>>(z, pei, pei + EP, out, EP, OUTC);
  decode_kernel<<<(EP + T - 1) / T, T, 0, stream>>>(z, nei, nei + EP, out + EP, EP, OUTC);
}